// YieldGNN_30897994728283
// MI455X (gfx1250) — compile-verified
//
#include <hip/hip_runtime.h>
#include <hip/hip_bf16.h>

#define N_NODES  100000
#define N_EDGES  1600000
#define N_FEAT   16
#define HIDDEN   128
#define N_GRAPHS 512

typedef float v2f __attribute__((ext_vector_type(2)));
typedef float v8f __attribute__((ext_vector_type(8)));

// ---------------------------------------------------------------- utilities

__global__ void fill_f32(float* __restrict__ p, float v, int n) {
    int i = blockIdx.x * blockDim.x + threadIdx.x;
    if (i < n) p[i] = v;
}

// deg[dst] += 1 per edge (deg pre-filled with 1.0 for the self-loop)
__global__ void edge_degree(const int* __restrict__ dst, float* __restrict__ deg, int nE) {
    int e = blockIdx.x * blockDim.x + threadIdx.x;
    if (e < nE) atomicAdd(&deg[dst[e]], 1.0f);
}

__global__ void deg_to_dinv(float* __restrict__ d, int n) {
    int i = blockIdx.x * blockDim.x + threadIdx.x;
    if (i < n) d[i] = rsqrtf(d[i]);
}

__global__ void edge_norm(const float* __restrict__ dinv, const int* __restrict__ src,
                          const int* __restrict__ dst, float* __restrict__ nrm, int nE) {
    int e = blockIdx.x * blockDim.x + threadIdx.x;
    if (e < nE) nrm[e] = dinv[src[e]] * dinv[dst[e]];
}

// ------------------------------------------------------- fp32 WMMA GEMM
// C[M x 128] = A[M x K] * B[K x 128], row-major, K multiple of 4, M multiple of 16.
// One wave -> one 16x16 tile of C via V_WMMA_F32_16X16X4_F32.
// blockDim = (32,4): 4 waves cover 4 adjacent N-tiles; gridDim = (M/16, 128/(16*4)).
template <int K>
__global__ void gemm_wmma_f32(const float* __restrict__ A, const float* __restrict__ B,
                              float* __restrict__ C) {
    const int m0   = blockIdx.x * 16;
    const int n0   = (blockIdx.y * 4 + threadIdx.y) * 16;
    const int lane = threadIdx.x;          // 0..31, wave32
    const int half = lane >> 4;            // 0: K pair {0,1}, 1: K pair {2,3}
    const int l15  = lane & 15;            // row (A) / col (B,C) within tile

    const float* __restrict__ arow = A + (size_t)(m0 + l15) * K;
    v8f acc = {};
#pragma unroll
    for (int k = 0; k < K; k += 4) {
        const int ka = k + half * 2;
        v2f a, b;
        a.x = arow[ka];
        a.y = arow[ka + 1];
        b.x = B[(size_t)(ka)     * HIDDEN + n0 + l15];
        b.y = B[(size_t)(ka + 1) * HIDDEN + n0 + l15];
        // (neg_a, A, neg_b, B, c_mod, C, reuse_a, reuse_b)
        acc = __builtin_amdgcn_wmma_f32_16x16x4_f32(false, a, false, b,
                                                    (short)0, acc, false, false);
    }
    // C/D layout: VGPR r -> M=r (lanes 0-15), M=r+8 (lanes 16-31)
    float* __restrict__ crow = C + (size_t)(m0 + half * 8) * HIDDEN + n0 + l15;
#pragma unroll
    for (int r = 0; r < 8; ++r) crow[(size_t)r * HIDDEN] = acc[r];
}

// -------------------------------------------------- edge scatter-aggregate
// 32 threads per edge, 4 features each: out[dst] += h[src] * norm[e]
__global__ void edge_aggregate(const float* __restrict__ h, const float* __restrict__ nrm,
                               const int* __restrict__ src, const int* __restrict__ dst,
                               float* __restrict__ out, int nE) {
    int t = blockIdx.x * blockDim.x + threadIdx.x;
    int e = t >> 5;
    if (e >= nE) return;
    int f  = (t & 31) * 4;
    int s  = src[e], d = dst[e];
    float w = nrm[e];
    const float4 hv = *(const float4*)(h + (size_t)s * HIDDEN + f);
    float* o = out + (size_t)d * HIDDEN + f;
    atomicAdd(o + 0, hv.x * w);
    atomicAdd(o + 1, hv.y * w);
    atomicAdd(o + 2, hv.z * w);
    atomicAdd(o + 3, hv.w * w);
}

// out = relu(out + h * dinv^2 + bias)
__global__ void finish_layer(float* __restrict__ out, const float* __restrict__ h,
                             const float* __restrict__ dinv, const float* __restrict__ bias,
                             int nTot) {
    int t = blockIdx.x * blockDim.x + threadIdx.x;
    if (t >= nTot) return;
    int n = t >> 7, f = t & (HIDDEN - 1);
    float di = dinv[n];
    float v  = out[t] + h[t] * di * di + bias[f];
    out[t] = fmaxf(v, 0.0f);
}

// -------------------------------------------------------------- pooling
__global__ void pool_scatter(const float* __restrict__ h, const int* __restrict__ batch,
                             float* __restrict__ sums, float* __restrict__ cnts, int nTot) {
    int t = blockIdx.x * blockDim.x + threadIdx.x;
    if (t >= nTot) return;
    int n = t >> 7, f = t & (HIDDEN - 1);
    int g = batch[n];
    atomicAdd(&sums[(size_t)g * HIDDEN + f], h[t]);
    if (f == 0) atomicAdd(&cnts[g], 1.0f);
}

// one wave per graph: out[g] = (sums[g]/max(cnt,1)) . Wl + bl
__global__ void readout(const float* __restrict__ sums, const float* __restrict__ cnts,
                        const float* __restrict__ Wl, const float* __restrict__ bl,
                        float* __restrict__ out, int nG) {
    int g    = (blockIdx.x * blockDim.x + threadIdx.x) >> 5;
    int lane = threadIdx.x & 31;
    if (g >= nG) return;
    float acc = 0.0f;
#pragma unroll
    for (int f = lane; f < HIDDEN; f += 32)
        acc += sums[(size_t)g * HIDDEN + f] * Wl[f];
#pragma unroll
    for (int off = 16; off; off >>= 1)
        acc += __shfl_down(acc, off, 32);
    if (lane == 0) {
        float inv = 1.0f / fmaxf(cnts[g], 1.0f);
        out[g] = acc * inv + bl[0];
    }
}

// ---------------------------------------------------------------- driver

static inline int cdiv(int a, int b) { return (a + b - 1) / b; }

extern "C" void kernel_launch(void* const* d_in, const int* in_sizes, int n_in,
                              void* d_out, int out_size, void* d_ws, size_t ws_size,
                              hipStream_t stream) {
    (void)in_sizes; (void)n_in; (void)out_size; (void)ws_size;

    const float* x     = (const float*)d_in[0];
    const int*   src   = (const int*)d_in[1];             // edge_index[0]
    const int*   dst   = ((const int*)d_in[1]) + N_EDGES; // edge_index[1]
    const int*   batch = (const int*)d_in[2];
    const float* W1    = (const float*)d_in[3];
    const float* b1    = (const float*)d_in[4];
    const float* W2    = (const float*)d_in[5];
    const float* b2    = (const float*)d_in[6];
    const float* Wl    = (const float*)d_in[7];
    const float* bl    = (const float*)d_in[8];
    float*       y     = (float*)d_out;

    const size_t NF = (size_t)N_NODES * HIDDEN;
    float* bufA   = (float*)d_ws;          // h (pre-aggregation features)
    float* bufB   = bufA + NF;             // aggregated output
    float* dinv   = bufB + NF;             // degree, then rsqrt(degree)
    float* nrm    = dinv + N_NODES;        // per-edge norm
    float* pooled = nrm + N_EDGES;         // [G, H] sums
    float* cnts   = pooled + (size_t)N_GRAPHS * HIDDEN;

    const int NT = N_NODES * HIDDEN;       // 12.8M
    const int ET = N_EDGES * 32;           // 51.2M feature-threads

    // --- degrees / norms ---
    fill_f32<<<cdiv(N_NODES, 256), 256, 0, stream>>>(dinv, 1.0f, N_NODES);
    edge_degree<<<cdiv(N_EDGES, 256), 256, 0, stream>>>(dst, dinv, N_EDGES);
    deg_to_dinv<<<cdiv(N_NODES, 256), 256, 0, stream>>>(dinv, N_NODES);
    edge_norm<<<cdiv(N_EDGES, 256), 256, 0, stream>>>(dinv, src, dst, nrm, N_EDGES);

    dim3 gblk(32, 4);
    dim3 ggrd(N_NODES / 16, HIDDEN / 64);

    // --- layer 1: h = x @ W1 ; aggregate ; + self-loop + bias ; relu ---
    gemm_wmma_f32<N_FEAT><<<ggrd, gblk, 0, stream>>>(x, W1, bufA);
    fill_f32<<<cdiv(NT, 256), 256, 0, stream>>>(bufB, 0.0f, NT);
    edge_aggregate<<<cdiv(ET, 256), 256, 0, stream>>>(bufA, nrm, src, dst, bufB, N_EDGES);
    finish_layer<<<cdiv(NT, 256), 256, 0, stream>>>(bufB, bufA, dinv, b1, NT);

    // --- layer 2: h = relu1 @ W2 ; aggregate ; + self-loop + bias ; relu ---
    gemm_wmma_f32<HIDDEN><<<ggrd, gblk, 0, stream>>>(bufB, W2, bufA);
    fill_f32<<<cdiv(NT, 256), 256, 0, stream>>>(bufB, 0.0f, NT);
    edge_aggregate<<<cdiv(ET, 256), 256, 0, stream>>>(bufA, nrm, src, dst, bufB, N_EDGES);
    finish_layer<<<cdiv(NT, 256), 256, 0, stream>>>(bufB, bufA, dinv, b2, NT);

    // --- mean pool + linear readout ---
    fill_f32<<<cdiv(N_GRAPHS * HIDDEN, 256), 256, 0, stream>>>(pooled, 0.0f, N_GRAPHS * HIDDEN);
    fill_f32<<<cdiv(N_GRAPHS, 256), 256, 0, stream>>>(cnts, 0.0f, N_GRAPHS);
    pool_scatter<<<cdiv(NT, 256), 256, 0, stream>>>(bufB, batch, pooled, cnts, NT);
    readout<<<cdiv(N_GRAPHS * 32, 256), 256, 0, stream>>>(pooled, cnts, Wl, bl, y, N_GRAPHS);
}